// MambaBlock_26869315404265
// MI455X (gfx1250) — compile-verified
//
#include <hip/hip_runtime.h>
#include <hip/hip_bf16.h>

// ---------------- problem constants (from reference) ----------------
#define DIMX       1024
#define D_STATE    64
#define D_CONV     4
#define HEADDIM    64
#define D_INNER    2048                      // EXPAND * DIM
#define NHEADS     32                        // D_INNER / HEADDIM
#define D_CONV_DIM 2176                      // D_INNER + 2*D_STATE
#define D_IN_PROJ  4256                      // 2*D_INNER + 2*D_STATE + NHEADS
#define BATCH      4
#define SEQ        2048
#define BL         (BATCH * SEQ)             // 8192 rows

typedef __attribute__((ext_vector_type(2))) float v2f;
typedef __attribute__((ext_vector_type(8))) float v8f;

// CDNA5 async global->LDS copy (ASYNCcnt-tracked, bypasses VGPRs).
// lds_off: per-lane LDS byte offset (low 32 bits of generic shared addr).
// gaddr:   per-lane 64-bit global address (16B aligned).
__device__ __forceinline__ void async_ld_b128(unsigned lds_off, const void* gaddr) {
    asm volatile("global_load_async_to_lds_b128 %0, %1, off"
                 :: "v"(lds_off), "v"(gaddr)
                 : "memory");
}

// =====================================================================
// Kernel 1/5: fp32 WMMA GEMM with async-LDS double buffering.
//   C[m][n] = sum_k A[m][k] * Bw[n][k]
//   A: row-major (M,K); Bw: row-major (N,K)  (native layout of both
//   in_proj_w and out_proj_w); C: row-major (M,N).
// Block: 256 threads = 8 waves; tile 128(M) x 64(N); K tiled by 32.
// Each wave computes a 16x64 strip = 4 accumulators of v_wmma_f32_16x16x4_f32.
// Next K-tile's global->LDS copies are in flight while WMMAs consume
// the current tile (s_wait_asynccnt pipelining).
// =====================================================================
#define TM 128
#define TN 64
#define TK 32
#define LSTR 36   // LDS row stride in floats: 144B = 16B aligned, conflict-free
#define A_TILE (TM * LSTR)
#define B_TILE (TN * LSTR)

__global__ __launch_bounds__(256) void gemm_wmma_f32(
    const float* __restrict__ A, const float* __restrict__ Bw,
    float* __restrict__ C, int M, int N, int K)
{
    __shared__ float sA[2 * A_TILE];
    __shared__ float sB[2 * B_TILE];

    const int m0   = blockIdx.x * TM;        // M assumed multiple of 128
    const int n0   = blockIdx.y * TN;        // N guarded at store
    const int tid  = threadIdx.x;
    const int wave = tid >> 5;               // 0..7
    const int lane = tid & 31;
    const int lh   = lane & 15;              // lane % 16
    const int khi  = (lane >> 4) << 1;       // 0 for lanes 0-15, 2 for 16-31

    // Per-thread copy coordinates (fixed across K-tiles).
    // A tile: 128x32 floats = 1024 b128 chunks -> 4 per thread.
    // B tile:  64x32 floats =  512 b128 chunks -> 2 per thread.
    const int rowA[4] = { (tid + 0*256) >> 3, (tid + 1*256) >> 3,
                          (tid + 2*256) >> 3, (tid + 3*256) >> 3 };
    const int c4     = (tid & 7) << 2;
    const int rowB0  = rowA[0];
    const int rowB1  = rowA[1];
    // clamp OOB B rows (garbage is fine; stores are guarded) so every
    // wave issues a uniform async count -> s_wait_asynccnt is exact.
    const int bgr0 = (n0 + rowB0 < N) ? (n0 + rowB0) : (N - 1);
    const int bgr1 = (n0 + rowB1 < N) ? (n0 + rowB1) : (N - 1);

    auto issue_tile = [&](int k0, int buf) {
        #pragma unroll
        for (int i = 0; i < 4; ++i) {
            async_ld_b128(
                (unsigned)(size_t)&sA[buf * A_TILE + rowA[i] * LSTR + c4],
                &A[(size_t)(m0 + rowA[i]) * K + (k0 + c4)]);
        }
        async_ld_b128((unsigned)(size_t)&sB[buf * B_TILE + rowB0 * LSTR + c4],
                      &Bw[(size_t)bgr0 * K + (k0 + c4)]);
        async_ld_b128((unsigned)(size_t)&sB[buf * B_TILE + rowB1 * LSTR + c4],
                      &Bw[(size_t)bgr1 * K + (k0 + c4)]);
    };

    v8f acc[4] = {};

    issue_tile(0, 0);                        // prologue: fill buffer 0

    for (int k0 = 0; k0 < K; k0 += TK) {
        const int buf = (k0 / TK) & 1;
        if (k0 + TK < K) {
            issue_tile(k0 + TK, buf ^ 1);    // prefetch next tile
            // 6 in flight for next tile; <=6 means current tile landed
            asm volatile("s_wait_asynccnt 0x6" ::: "memory");
        } else {
            asm volatile("s_wait_asynccnt 0x0" ::: "memory");
        }
        __syncthreads();                     // current tile visible to all waves

        // ---- 8 k-steps of K=4 WMMA; A frag reused across 4 N-tiles ----
        const float* pa = &sA[buf * A_TILE + (wave * 16 + lh) * LSTR + khi];
        const float* pb = &sB[buf * B_TILE];
        #pragma unroll
        for (int kk = 0; kk < 8; ++kk) {
            // A 16x4 layout: lanes 0-15 M=0..15 K={0,1}; lanes 16-31 K={2,3}
            v2f afrag = *reinterpret_cast<const v2f*>(pa + kk * 4);
            #pragma unroll
            for (int nt = 0; nt < 4; ++nt) {
                // B 4x16 layout: lanes 0-15 N=0..15 K={0,1}; lanes 16-31 K={2,3}
                v2f bfrag = *reinterpret_cast<const v2f*>(
                    &pb[(nt * 16 + lh) * LSTR + kk * 4 + khi]);
                acc[nt] = __builtin_amdgcn_wmma_f32_16x16x4_f32(
                    /*neg_a=*/false, afrag, /*neg_b=*/false, bfrag,
                    /*c_mod=*/(short)0, acc[nt],
                    /*reuse_a=*/false, /*reuse_b=*/false);
            }
        }
        __syncthreads();                     // don't let next issue race readers
    }

    // ---- store: VGPR r holds (M = r + 8*(lane>=16), N = lane%16 + 16*nt) ----
    const int mrow = m0 + wave * 16 + (lane >> 4) * 8;
    #pragma unroll
    for (int nt = 0; nt < 4; ++nt) {
        int ncol = n0 + nt * 16 + lh;
        if (ncol < N) {
            #pragma unroll
            for (int r = 0; r < 8; ++r) {
                C[(size_t)(mrow + r) * N + ncol] = acc[nt][r];
            }
        }
    }
}

// =====================================================================
// Kernel 2: depthwise conv(k=4) + bias + SiLU on xBC channels, and
// softplus(dt_raw + dt_bias) for the 32 dt channels. One thread per
// (b, l, channel) over 2176 + 32 = 2208 channels.
// =====================================================================
__global__ __launch_bounds__(256) void conv_silu_dt_kernel(
    const float* __restrict__ zxbcdt,   // (BL, 4256)
    const float* __restrict__ conv_w,   // (2176, 4)
    const float* __restrict__ conv_b,   // (2176,)
    const float* __restrict__ dt_bias,  // (32,)
    float* __restrict__ xbc_out,        // (BL, 2176)
    float* __restrict__ dt_out)         // (BL, 32)
{
    const int CH = D_CONV_DIM + NHEADS;              // 2208
    long idx = (long)blockIdx.x * blockDim.x + threadIdx.x;
    if (idx >= (long)BL * CH) return;
    int  c  = (int)(idx % CH);
    long bl = idx / CH;
    int  l  = (int)(bl % SEQ);

    if (c < D_CONV_DIM) {
        float acc = conv_b[c];
        #pragma unroll
        for (int k = 0; k < D_CONV; ++k) {
            int ls = l + k - (D_CONV - 1);           // causal, zero-padded
            if (ls >= 0) {
                acc += zxbcdt[(bl + (long)(ls - l)) * D_IN_PROJ + D_INNER + c]
                     * conv_w[c * D_CONV + k];
            }
        }
        float s = acc / (1.0f + __expf(-acc));       // SiLU
        xbc_out[bl * D_CONV_DIM + c] = s;
    } else {
        int h = c - D_CONV_DIM;
        float v = zxbcdt[bl * D_IN_PROJ + (D_INNER + D_CONV_DIM) + h] + dt_bias[h];
        float sp = (v > 20.0f) ? v : __logf(1.0f + __expf(v));  // softplus
        dt_out[bl * NHEADS + h] = sp;
    }
}

// =====================================================================
// Kernel 3: sequential SSM scan. One 64-thread block per (batch, head).
// Thread p owns row h[p, 0..63] of the state entirely in registers;
// B_t / C_t broadcast through LDS each step. y_p = dot(h[p,:], C_t).
// =====================================================================
__global__ __launch_bounds__(64) void scan_kernel(
    const float* __restrict__ xbc,   // (BL, 2176): [x_ssm | B | C]
    const float* __restrict__ dt,    // (BL, 32)
    const float* __restrict__ A_log, // (32,)
    const float* __restrict__ Dp,    // (32,)
    float* __restrict__ y)           // (BL, 2048)
{
    const int b    = blockIdx.x / NHEADS;
    const int head = blockIdx.x % NHEADS;
    const int p    = threadIdx.x;                    // 0..63
    const float Ah = -__expf(A_log[head]);
    const float Dh = Dp[head];

    float h[D_STATE];
    #pragma unroll
    for (int n = 0; n < D_STATE; ++n) h[n] = 0.0f;

    __shared__ float sB[D_STATE];
    __shared__ float sC[D_STATE];
    __shared__ float sdt;

    for (int l = 0; l < SEQ; ++l) {
        const long rowi = (long)b * SEQ + l;
        const float* rx = xbc + rowi * D_CONV_DIM;
        sB[p] = rx[D_INNER + p];
        sC[p] = rx[D_INNER + D_STATE + p];
        if (p == 0) sdt = dt[rowi * NHEADS + head];
        const float xt = rx[head * HEADDIM + p];
        __syncthreads();

        const float dtt  = sdt;
        const float dA   = __expf(dtt * Ah);
        const float coef = dtt * xt;
        float acc = 0.0f;
        #pragma unroll
        for (int n = 0; n < D_STATE; ++n) {
            h[n] = h[n] * dA + coef * sB[n];
            acc += h[n] * sC[n];
        }
        y[rowi * D_INNER + head * HEADDIM + p] = acc + Dh * xt;
        __syncthreads();                             // protect sB/sC reuse
    }
}

// =====================================================================
// Kernel 4: g = y * silu(z); RMSNorm over the 2048 channels; * norm_w.
// One 256-thread block per (b,l) row; written in-place over y.
// =====================================================================
__global__ __launch_bounds__(256) void gate_norm_kernel(
    const float* __restrict__ zxbcdt,  // z = first 2048 of each row
    float* __restrict__ ybuf,          // (BL, 2048), in/out
    const float* __restrict__ norm_w)
{
    const long row = blockIdx.x;
    const float* z  = zxbcdt + row * D_IN_PROJ;
    float*       yr = ybuf + row * D_INNER;

    float g[8];
    float ss = 0.0f;
    #pragma unroll
    for (int i = 0; i < 8; ++i) {
        int e = threadIdx.x + i * 256;
        float zz = z[e];
        float gg = yr[e] * (zz / (1.0f + __expf(-zz)));
        g[i] = gg;
        ss += gg * gg;
    }
    __shared__ float red[256];
    red[threadIdx.x] = ss;
    __syncthreads();
    #pragma unroll
    for (int s = 128; s > 0; s >>= 1) {
        if (threadIdx.x < s) red[threadIdx.x] += red[threadIdx.x + s];
        __syncthreads();
    }
    const float scale = rsqrtf(red[0] * (1.0f / D_INNER) + 1e-5f);
    #pragma unroll
    for (int i = 0; i < 8; ++i) {
        int e = threadIdx.x + i * 256;
        yr[e] = g[i] * scale * norm_w[e];
    }
}

// =====================================================================
extern "C" void kernel_launch(void* const* d_in, const int* in_sizes, int n_in,
                              void* d_out, int out_size, void* d_ws, size_t ws_size,
                              hipStream_t stream) {
    const float* x          = (const float*)d_in[0];
    const float* in_proj_w  = (const float*)d_in[1];
    const float* conv_w     = (const float*)d_in[2];
    const float* conv_b     = (const float*)d_in[3];
    const float* dt_bias    = (const float*)d_in[4];
    const float* A_log      = (const float*)d_in[5];
    const float* Dp         = (const float*)d_in[6];
    const float* norm_w     = (const float*)d_in[7];
    const float* out_proj_w = (const float*)d_in[8];
    float*       out        = (float*)d_out;

    // workspace partition (floats)
    float* zxbcdt = (float*)d_ws;                              // BL*4256
    float* xbc    = zxbcdt + (size_t)BL * D_IN_PROJ;           // BL*2176
    float* dtb    = xbc    + (size_t)BL * D_CONV_DIM;          // BL*32
    float* ybuf   = dtb    + (size_t)BL * NHEADS;              // BL*2048

    // 1) in_proj GEMM: (8192,1024) x (4256,1024)^T -> (8192,4256)
    {
        dim3 grid(BL / TM, (D_IN_PROJ + TN - 1) / TN);         // 64 x 67
        gemm_wmma_f32<<<grid, 256, 0, stream>>>(x, in_proj_w, zxbcdt,
                                                BL, D_IN_PROJ, DIMX);
    }
    // 2) conv + SiLU + softplus(dt)
    {
        long total = (long)BL * (D_CONV_DIM + NHEADS);
        int blocks = (int)((total + 255) / 256);
        conv_silu_dt_kernel<<<blocks, 256, 0, stream>>>(
            zxbcdt, conv_w, conv_b, dt_bias, xbc, dtb);
    }
    // 3) SSM scan: one block per (batch, head)
    scan_kernel<<<BATCH * NHEADS, 64, 0, stream>>>(xbc, dtb, A_log, Dp, ybuf);
    // 4) gate + RMSNorm (in place on ybuf)
    gate_norm_kernel<<<BL, 256, 0, stream>>>(zxbcdt, ybuf, norm_w);
    // 5) out_proj GEMM: (8192,2048) x (1024,2048)^T -> (8192,1024)
    {
        dim3 grid(BL / TM, DIMX / TN);                          // 64 x 16
        gemm_wmma_f32<<<grid, 256, 0, stream>>>(ybuf, out_proj_w, out,
                                                BL, DIMX, D_INNER);
    }
}